// SelfAttentionBlock_9234179686733
// MI455X (gfx1250) — compile-verified
//
#include <hip/hip_runtime.h>
#include <math.h>

typedef __attribute__((ext_vector_type(2))) float v2f;
typedef __attribute__((ext_vector_type(8))) float v8f;

#define DIM      512
#define HEADS    8
#define HEAD_DIM 64
#define NTOK     1024       // 32*32
#define BATCH    8
#define QKV_DIM  1536
#define ATT_SCALE 0.125f    // 64^-0.5
#define LN_EPS    1e-5f

// D = A(16x4 f32) * B(4x16 f32) + C(16x16 f32), exact f32 matrix op on CDNA5.
static __device__ __forceinline__ v8f wmma4(v2f a, v2f b, v8f c) {
  return __builtin_amdgcn_wmma_f32_16x16x4_f32(false, a, false, b, (short)0, c,
                                               false, false);
}

// ---------------------------------------------------------------------------
// Kernel 1: LayerNorm over channels; [b,c,h,w] -> x_norm [b*n, c] contiguous.
// ---------------------------------------------------------------------------
__global__ void ln_kernel(const float* __restrict__ x,
                          const float* __restrict__ gamma,
                          const float* __restrict__ beta,
                          float* __restrict__ xn) {
  const int row = blockIdx.x;            // b*NTOK + pos
  const int b   = row >> 10;
  const int pos = row & (NTOK - 1);
  const int tid = threadIdx.x;           // 256 threads, 2 channels each

  const float* xp = x + (size_t)b * DIM * NTOK + pos;
  const float v0 = xp[(size_t)tid * NTOK];
  const float v1 = xp[(size_t)(tid + 256) * NTOK];

  float s  = v0 + v1;
  float s2 = v0 * v0 + v1 * v1;
  #pragma unroll
  for (int m = 16; m; m >>= 1) {
    s  += __shfl_xor(s,  m, 32);
    s2 += __shfl_xor(s2, m, 32);
  }
  __shared__ float rs[8], rs2[8];
  const int wv = tid >> 5, ln = tid & 31;
  if (ln == 0) { rs[wv] = s; rs2[wv] = s2; }
  __syncthreads();
  if (tid == 0) {
    float a = 0.f, c = 0.f;
    #pragma unroll
    for (int i = 0; i < 8; ++i) { a += rs[i]; c += rs2[i]; }
    rs[0] = a; rs2[0] = c;
  }
  __syncthreads();
  const float mean = rs[0] * (1.0f / DIM);
  const float var  = rs2[0] * (1.0f / DIM) - mean * mean;
  const float inv  = rsqrtf(var + LN_EPS);

  float* op = xn + (size_t)row * DIM;
  op[tid]       = (v0 - mean) * inv * gamma[tid]       + beta[tid];
  op[tid + 256] = (v1 - mean) * inv * gamma[tid + 256] + beta[tid + 256];
}

// ---------------------------------------------------------------------------
// Kernel 2: QKV GEMM.  qkv[m, d] = sum_k xn[m,k] * w[d,k]
// M = 8192, N = 1536, K = 512.  Wave tile: 16x64.  Block: 8 waves = 128x64.
// A frag: lane(M=lane&15) float2 @ k0+2*(lane>>4).  B frag: same, N=lane&15.
// ---------------------------------------------------------------------------
__global__ void qkv_gemm_kernel(const float* __restrict__ xn,
                                const float* __restrict__ w,
                                float* __restrict__ qkv) {
  const int lane = threadIdx.x & 31, wave = threadIdx.x >> 5;
  const int laneM = lane & 15, half = lane >> 4;
  const int m0 = blockIdx.x * 128 + wave * 16;
  const int n0 = blockIdx.y * 64;

  v8f acc0 = {}, acc1 = {}, acc2 = {}, acc3 = {};
  const float* ap = xn + (size_t)(m0 + laneM) * DIM + 2 * half;
  const float* bp = w  + (size_t)(n0 + laneM) * DIM + 2 * half;

  #pragma unroll 4
  for (int k0 = 0; k0 < DIM; k0 += 4) {
    const v2f a  = *(const v2f*)(ap + k0);
    const v2f b0 = *(const v2f*)(bp + k0);
    const v2f b1 = *(const v2f*)(bp + 16 * DIM + k0);
    const v2f b2 = *(const v2f*)(bp + 32 * DIM + k0);
    const v2f b3 = *(const v2f*)(bp + 48 * DIM + k0);
    acc0 = wmma4(a, b0, acc0);
    acc1 = wmma4(a, b1, acc1);
    acc2 = wmma4(a, b2, acc2);
    acc3 = wmma4(a, b3, acc3);
  }

  #pragma unroll
  for (int r = 0; r < 8; ++r) {
    const int m = m0 + r + 8 * half;            // C layout: VGPR r -> M=r / r+8
    float* op = qkv + (size_t)m * QKV_DIM + n0 + laneM;
    op[0]  = acc0[r];
    op[16] = acc1[r];
    op[32] = acc2[r];
    op[48] = acc3[r];
  }
}

// ---------------------------------------------------------------------------
// Kernel 3: flash attention per (b, head).  Block = 4 waves; each wave owns a
// 16-row query tile, loops over 64 key tiles of 16.  O accumulated in regs.
// ---------------------------------------------------------------------------
__global__ void attn_kernel(const float* __restrict__ qkv,
                            float* __restrict__ ao) {
  const int lane = threadIdx.x & 31, wave = threadIdx.x >> 5;
  const int laneM = lane & 15, half = lane >> 4;
  const int q0   = blockIdx.x * 64 + wave * 16;
  const int head = blockIdx.y;
  const int b    = blockIdx.z;

  const float* base  = qkv + (size_t)b * NTOK * QKV_DIM + head * HEAD_DIM;
  const float* kbase = base + DIM;       // K at offset 512
  const float* vbase = base + 2 * DIM;   // V at offset 1024

  // Preload Q (16x64 per wave) as A-fragments, pre-scaled by 1/sqrt(hd).
  v2f qf[16];
  {
    const float* qp = base + (size_t)(q0 + laneM) * QKV_DIM + 2 * half;
    #pragma unroll
    for (int ks = 0; ks < 16; ++ks) {
      v2f t = *(const v2f*)(qp + 4 * ks);
      qf[ks] = t * ATT_SCALE;
    }
  }

  v8f acc[4] = {};                   // O tile: 16 rows x 64 cols
  float rmax[8], rsum[8];
  #pragma unroll
  for (int r = 0; r < 8; ++r) { rmax[r] = -3.0e38f; rsum[r] = 0.0f; }

  __shared__ float plds[4][16][16];  // per-wave P tile (C-layout -> A-layout)

  for (int j0 = 0; j0 < NTOK; j0 += 16) {
    // ---- S = Q * K^T (scaled) : 16 WMMAs ----
    v8f s = {};
    const float* kp = kbase + (size_t)(j0 + laneM) * QKV_DIM + 2 * half;
    #pragma unroll
    for (int ks = 0; ks < 16; ++ks) {
      const v2f bf = *(const v2f*)(kp + 4 * ks);
      s = wmma4(qf[ks], bf, s);
    }

    // ---- online softmax: per-row (r + 8*half) max/sum across 16-lane halves
    float alpha[8];
    #pragma unroll
    for (int r = 0; r < 8; ++r) {
      float v = s[r];
      v = fmaxf(v, __shfl_xor(v, 1, 32));
      v = fmaxf(v, __shfl_xor(v, 2, 32));
      v = fmaxf(v, __shfl_xor(v, 4, 32));
      v = fmaxf(v, __shfl_xor(v, 8, 32));
      const float nm = fmaxf(rmax[r], v);
      alpha[r] = __expf(rmax[r] - nm);
      rmax[r] = nm;
      const float p = __expf(s[r] - nm);
      s[r] = p;
      float ts = p;
      ts += __shfl_xor(ts, 1, 32);
      ts += __shfl_xor(ts, 2, 32);
      ts += __shfl_xor(ts, 4, 32);
      ts += __shfl_xor(ts, 8, 32);
      rsum[r] = rsum[r] * alpha[r] + ts;
    }

    // ---- rescale running O ----
    #pragma unroll
    for (int t = 0; t < 4; ++t)
      #pragma unroll
      for (int r = 0; r < 8; ++r)
        acc[t][r] *= alpha[r];

    // ---- stage P through LDS to get A-layout fragments ----
    #pragma unroll
    for (int r = 0; r < 8; ++r)
      plds[wave][r + 8 * half][laneM] = s[r];
    __syncthreads();

    // ---- O += P * V : 16 WMMAs ----
    #pragma unroll
    for (int ks = 0; ks < 4; ++ks) {
      v2f af;
      af.x = plds[wave][laneM][4 * ks + 2 * half];
      af.y = plds[wave][laneM][4 * ks + 2 * half + 1];
      const float* vp = vbase + (size_t)(j0 + 4 * ks + 2 * half) * QKV_DIM + laneM;
      #pragma unroll
      for (int t = 0; t < 4; ++t) {
        v2f bf;
        bf.x = vp[t * 16];
        bf.y = vp[QKV_DIM + t * 16];
        acc[t] = wmma4(af, bf, acc[t]);
      }
    }
    __syncthreads();
  }

  // ---- epilogue: normalize and write [b*n, c] ----
  #pragma unroll
  for (int t = 0; t < 4; ++t) {
    #pragma unroll
    for (int r = 0; r < 8; ++r) {
      const int m = q0 + r + 8 * half;
      const int c = head * HEAD_DIM + t * 16 + laneM;
      ao[((size_t)b * NTOK + m) * DIM + c] = acc[t][r] / rsum[r];
    }
  }
}

// ---------------------------------------------------------------------------
// Kernel 4: output projection + bias + residual, write back [b,c,h,w].
// out[b,c,pos] = sum_k ao[m,k]*w[c,k] + bias[c] + x[b,c,pos],  m = b*n+pos
// ---------------------------------------------------------------------------
__global__ void proj_kernel(const float* __restrict__ ao,
                            const float* __restrict__ w,
                            const float* __restrict__ bias,
                            const float* __restrict__ x,
                            float* __restrict__ out) {
  const int lane = threadIdx.x & 31, wave = threadIdx.x >> 5;
  const int laneM = lane & 15, half = lane >> 4;
  const int m0 = blockIdx.x * 128 + wave * 16;
  const int n0 = blockIdx.y * 64;

  v8f acc0 = {}, acc1 = {}, acc2 = {}, acc3 = {};
  const float* ap = ao + (size_t)(m0 + laneM) * DIM + 2 * half;
  const float* bp = w  + (size_t)(n0 + laneM) * DIM + 2 * half;

  #pragma unroll 4
  for (int k0 = 0; k0 < DIM; k0 += 4) {
    const v2f a  = *(const v2f*)(ap + k0);
    const v2f b0 = *(const v2f*)(bp + k0);
    const v2f b1 = *(const v2f*)(bp + 16 * DIM + k0);
    const v2f b2 = *(const v2f*)(bp + 32 * DIM + k0);
    const v2f b3 = *(const v2f*)(bp + 48 * DIM + k0);
    acc0 = wmma4(a, b0, acc0);
    acc1 = wmma4(a, b1, acc1);
    acc2 = wmma4(a, b2, acc2);
    acc3 = wmma4(a, b3, acc3);
  }

  #pragma unroll
  for (int r = 0; r < 8; ++r) {
    const int m   = m0 + r + 8 * half;
    const int b   = m >> 10;
    const int pos = m & (NTOK - 1);
    #pragma unroll
    for (int t = 0; t < 4; ++t) {
      const int c = n0 + t * 16 + laneM;
      const size_t oidx = ((size_t)b * DIM + c) * NTOK + pos;
      const float v = (t == 0 ? acc0[r] : t == 1 ? acc1[r] : t == 2 ? acc2[r] : acc3[r]);
      out[oidx] = v + bias[c] + x[oidx];
    }
  }
}

// ---------------------------------------------------------------------------
extern "C" void kernel_launch(void* const* d_in, const int* in_sizes, int n_in,
                              void* d_out, int out_size, void* d_ws, size_t ws_size,
                              hipStream_t stream) {
  const float* x      = (const float*)d_in[0];
  const float* gamma  = (const float*)d_in[1];
  const float* beta   = (const float*)d_in[2];
  const float* w_qkv  = (const float*)d_in[3];
  const float* w_proj = (const float*)d_in[4];
  const float* b_proj = (const float*)d_in[5];
  float* out = (float*)d_out;

  float* xn  = (float*)d_ws;                                  // 8192*512  f32
  float* qkv = xn  + (size_t)BATCH * NTOK * DIM;              // 8192*1536 f32
  float* ao  = qkv + (size_t)BATCH * NTOK * QKV_DIM;          // 8192*512  f32

  ln_kernel<<<dim3(BATCH * NTOK), dim3(256), 0, stream>>>(x, gamma, beta, xn);
  qkv_gemm_kernel<<<dim3(64, QKV_DIM / 64), dim3(256), 0, stream>>>(xn, w_qkv, qkv);
  attn_kernel<<<dim3(NTOK / 64, HEADS, BATCH), dim3(128), 0, stream>>>(qkv, ao);
  proj_kernel<<<dim3(64, DIM / 64), dim3(256), 0, stream>>>(ao, w_proj, b_proj, x, out);
}